// SparseGrid_67276367725168
// MI455X (gfx1250) — compile-verified
//
#include <hip/hip_runtime.h>

// ---------------------------------------------------------------------------
// Plenoxels (svox2 "cuvol") volume renderer for MI455X (gfx1250, wave32).
// One wave per ray; lanes = 32 consecutive ray-march steps per iteration.
// Gather-bound (L2-resident on 192MB L2); WMMA inapplicable (no shared
// operand matrix). CDNA5 paths: async Global->LDS double-buffering of the
// density gather (global_load_async_to_lds_b32 + s_wait_asynccnt),
// ds_bpermute wave32 prefix scan, global_prefetch_b8 one batch ahead.
// Cell base/fractions are computed once per batch and rotated through
// registers (used for async issue at b, reused for the trilerp at b+1).
// ---------------------------------------------------------------------------

#define RESO        128
#define NSTEPS      448            // ceil(sqrt(3)*128/0.5)+4
#define NB          (NSTEPS/32)    // 14 batches of 32 steps
#define STEPSZ      0.5f
#define SIGMA_TH    1e-10f
#define STOP_TH     1e-7f
#define LOG_STOP    (-16.11809565095832f)   // ln(1e-7)
#define POS_CLAMP   126.9999f               // RESO-1-1e-4
#define XSTRIDE     (RESO*RESO)             // 16384

__device__ __forceinline__ float bperm(int src_lane, float v) {
    // dst[lane] = src[src_lane]  (index in bytes; bits [6:2] used, wraps)
    return __int_as_float(__builtin_amdgcn_ds_bpermute(src_lane * 4, __float_as_int(v)));
}

__device__ __forceinline__ int imin(int a, int b) { return a < b ? a : b; }

// Voxel cell (base index + fractional weights) for ray-march step `step`.
__device__ __forceinline__ int batch_base(float ogx, float ogy, float ogz,
                                          float gx, float gy, float gz,
                                          float tmin, int step,
                                          float& fx, float& fy, float& fz) {
    const float t  = tmin + (float)step * STEPSZ;
    const float px = fminf(fmaxf(ogx + t*gx, 0.f), POS_CLAMP);
    const float py = fminf(fmaxf(ogy + t*gy, 0.f), POS_CLAMP);
    const float pz = fminf(fmaxf(ogz + t*gz, 0.f), POS_CLAMP);
    const int lx = imin((int)px, 126);
    const int ly = imin((int)py, 126);
    const int lz = imin((int)pz, 126);
    fx = px - (float)lx; fy = py - (float)ly; fz = pz - (float)lz;
    return (lx*RESO + ly)*RESO + lz;
}

// Issue 8 async B32 gathers (the 8 trilerp corners) into this lane's LDS
// slots. Tracked on ASYNCcnt; completion enforced by s_wait_asynccnt.
__device__ __forceinline__ void issue_density_async(const float* __restrict__ dens,
                                                    int base, float* lds_slot0) {
    const int offs[8] = {0, 1, RESO, RESO + 1,
                         XSTRIDE, XSTRIDE + 1, XSTRIDE + RESO, XSTRIDE + RESO + 1};
    #pragma unroll
    for (int c = 0; c < 8; ++c) {
        // Flat LDS pointers carry the LDS byte offset in addr[31:0].
        unsigned loff = (unsigned)(size_t)(lds_slot0 + c*32);
        const float* gaddr = dens + base + offs[c];
        asm volatile("global_load_async_to_lds_b32 %0, %1, off"
                     :: "v"(loff), "v"(gaddr) : "memory");
    }
}

__global__ __launch_bounds__(256) void plenoxel_render(
    const float* __restrict__ dens,   // [RESO^3, 1]
    const float* __restrict__ sh,     // [RESO^3, 27]
    const float* __restrict__ org,    // [2048, 3]
    const float* __restrict__ dir,    // [2048, 3]
    float*       __restrict__ out)    // [2048, 3]
{
    const int lane = threadIdx.x & 31;
    const int wv   = threadIdx.x >> 5;
    const int ray  = (blockIdx.x << 3) + wv;

    // Double-buffered per-wave density staging: [buf][wave][corner][lane]
    __shared__ float sdens[2][8][8][32];

    // ---------------- per-ray setup (uniform across the wave) --------------
    float dx = dir[ray*3+0], dy = dir[ray*3+1], dz = dir[ray*3+2];
    float rn = rsqrtf(dx*dx + dy*dy + dz*dz);
    dx *= rn; dy *= rn; dz *= rn;
    const float ox = org[ray*3+0], oy = org[ray*3+1], oz = org[ray*3+2];

    // world -> grid:  g = RESO*(p*0.5 + 0.5) - 0.5
    const float ogx = (float)RESO * (ox*0.5f + 0.5f) - 0.5f;
    const float ogy = (float)RESO * (oy*0.5f + 0.5f) - 0.5f;
    const float ogz = (float)RESO * (oz*0.5f + 0.5f) - 0.5f;
    float gx = dx * 64.f, gy = dy * 64.f, gz = dz * 64.f;
    const float delta_scale = rsqrtf(gx*gx + gy*gy + gz*gz);
    gx *= delta_scale; gy *= delta_scale; gz *= delta_scale;   // unit in grid space
    const float world_step = STEPSZ * delta_scale;

    // slab intersection with [0, RESO-1]^3
    const float sx = (fabsf(gx) < 1e-9f) ? 1e-9f : gx;
    const float sy = (fabsf(gy) < 1e-9f) ? 1e-9f : gy;
    const float sz = (fabsf(gz) < 1e-9f) ? 1e-9f : gz;
    const float t1x = (0.f - ogx)/sx, t2x = (127.f - ogx)/sx;
    const float t1y = (0.f - ogy)/sy, t2y = (127.f - ogy)/sy;
    const float t1z = (0.f - ogz)/sz, t2z = (127.f - ogz)/sz;
    const float tmin = fmaxf(fmaxf(fminf(t1x,t2x), fmaxf(fminf(t1y,t2y), fminf(t1z,t2z))), 0.f);
    const float tmax = fminf(fmaxf(t1x,t2x), fminf(fmaxf(t1y,t2y), fmaxf(t1z,t2z)));

    // SH degree-2 basis of the (unit) direction
    float shm[9];
    shm[0] = 0.28209479177387814f;
    shm[1] = -0.4886025119029199f * dy;
    shm[2] =  0.4886025119029199f * dz;
    shm[3] = -0.4886025119029199f * dx;
    shm[4] =  1.0925484305920792f * dx * dy;
    shm[5] = -1.0925484305920792f * dy * dz;
    shm[6] =  0.31539156525252005f * (2.f*dz*dz - dx*dx - dy*dy);
    shm[7] = -1.0925484305920792f * dx * dz;
    shm[8] =  0.5462742152960396f * (dx*dx - dy*dy);

    // Batch 0: compute cell, issue async density gather into buffer 0.
    float fx, fy, fz;
    int base = batch_base(ogx,ogy,ogz, gx,gy,gz, tmin, lane, fx, fy, fz);
    issue_density_async(dens, base, &sdens[0][wv][0][lane]);

    float carry = 0.f;                   // running log-transmittance
    float accR = 0.f, accG = 0.f, accB = 0.f;

    for (int b = 0; b < NB; ++b) {
        // ---- stage batch b+1: async density gather + SH prefetch ----------
        float nfx, nfy, nfz;
        const int base_n = batch_base(ogx,ogy,ogz, gx,gy,gz, tmin,
                                      (b+1)*32 + lane, nfx, nfy, nfz);
        issue_density_async(dens, base_n, &sdens[(b+1)&1][wv][0][lane]);
        {
            const float* p = sh + (size_t)base_n * 27;
            __builtin_prefetch(p, 0, 0);                       // global_prefetch_b8
            __builtin_prefetch(p + (size_t)XSTRIDE * 27, 0, 0);
        }
        // Async loads complete in order: <=8 outstanding => batch b's 8 landed.
        asm volatile("s_wait_asynccnt 0x8" ::: "memory");

        // ---- batch b: weights from the rotated cell state ----
        const float t = tmin + (float)(b*32 + lane) * STEPSZ;
        const bool valid = (t <= tmax);
        const float wx1 = fx, wx0 = 1.f - fx;
        const float wy1 = fy, wy0 = 1.f - fy;
        const float wz1 = fz, wz0 = 1.f - fz;

        // ---- density trilerp from the LDS staging buffer ----
        const float* slot = &sdens[b&1][wv][0][lane];
        const float d000 = slot[0*32], d001 = slot[1*32];
        const float d010 = slot[2*32], d011 = slot[3*32];
        const float d100 = slot[4*32], d101 = slot[5*32];
        const float d110 = slot[6*32], d111 = slot[7*32];
        float sig = wx0*(wy0*(wz0*d000 + wz1*d001) + wy1*(wz0*d010 + wz1*d011))
                  + wx1*(wy0*(wz0*d100 + wz1*d101) + wy1*(wz0*d110 + wz1*d111));
        if (!valid || !(sig > SIGMA_TH)) sig = 0.f;
        const float la = -sig * world_step;      // log attenuation (<= 0)

        // ---- wave32 inclusive prefix scan of la (Kogge-Stone, ds_bpermute) --
        float scan = la;
        #pragma unroll
        for (int off = 1; off < 32; off <<= 1) {
            const float up = bperm(lane - off, scan);
            if (lane >= off) scan += up;
        }
        const float total = bperm(31, scan);     // batch total (broadcast)
        const float logT  = carry + (scan - la); // exclusive prefix
        const float T     = __expf(logT);
        float w = T * (1.f - __expf(la));
        if (T < STOP_TH) w = 0.f;
        carry += total;

        // ---- SH trilerp + color only when this step contributes ----
        if (w != 0.f) {
            const int   offs[4] = {0, RESO, XSTRIDE, XSTRIDE + RESO};
            const float wxy[4]  = {wx0*wy0, wx0*wy1, wx1*wy0, wx1*wy1};
            float cr = 0.f, cg = 0.f, cb = 0.f;
            #pragma unroll
            for (int j = 0; j < 4; ++j) {
                const float* row = sh + (size_t)(base + offs[j]) * 27;  // 54 contiguous floats
                float a0=0.f,a1=0.f,a2=0.f, b0=0.f,b1=0.f,b2=0.f;
                #pragma unroll
                for (int k = 0; k < 9; ++k) {
                    const float m = shm[k];
                    a0 += m * row[k];      a1 += m * row[9+k];  a2 += m * row[18+k];
                    b0 += m * row[27+k];   b1 += m * row[36+k]; b2 += m * row[45+k];
                }
                const float wA = wxy[j]*wz0, wB = wxy[j]*wz1;
                cr += wA*a0 + wB*b0;
                cg += wA*a1 + wB*b1;
                cb += wA*a2 + wB*b2;
            }
            cr = fmaxf(cr + 0.5f, 0.f);
            cg = fmaxf(cg + 0.5f, 0.f);
            cb = fmaxf(cb + 0.5f, 0.f);
            accR += w * cr;  accG += w * cg;  accB += w * cb;
        }

        // rotate cell state: batch b+1 becomes current
        base = base_n; fx = nfx; fy = nfy; fz = nfz;

        // uniform early exits (identical across the wave); outstanding async
        // loads are drained by the implicit wait-idle at s_endpgm.
        if (carry < LOG_STOP) break;                         // T below stop forever
        if (tmin + (float)((b+1)*32) * STEPSZ > tmax) break; // rest of steps invalid
    }

    float Tf = __expf(carry);
    if (Tf < STOP_TH) Tf = 0.f;

    // ---- reduce the 3 accumulators across the wave (xor butterfly) ----
    #pragma unroll
    for (int off = 16; off > 0; off >>= 1) {
        accR += bperm(lane ^ off, accR);
        accG += bperm(lane ^ off, accG);
        accB += bperm(lane ^ off, accB);
    }

    if (lane == 0) {
        out[ray*3+0] = accR + Tf;   // BG_BRIGHTNESS = 1.0
        out[ray*3+1] = accG + Tf;
        out[ray*3+2] = accB + Tf;
    }
}

extern "C" void kernel_launch(void* const* d_in, const int* in_sizes, int n_in,
                              void* d_out, int out_size, void* d_ws, size_t ws_size,
                              hipStream_t stream) {
    const float* dens = (const float*)d_in[0];   // [128^3, 1]
    const float* sh   = (const float*)d_in[1];   // [128^3, 27]
    const float* org  = (const float*)d_in[2];   // [2048, 3]
    const float* dirs = (const float*)d_in[3];   // [2048, 3]
    float* out = (float*)d_out;                  // [2048, 3]

    dim3 block(256);                             // 8 waves = 8 rays per block
    dim3 grid(2048 / 8);
    plenoxel_render<<<grid, block, 0, stream>>>(dens, sh, org, dirs, out);
}